// LABRBF_3547642987289
// MI455X (gfx1250) — compile-verified
//
#include <hip/hip_runtime.h>
#include <hip/hip_bf16.h>
#include <math.h>

#define F    32
#define S    1024
#define N    8192
#define LAMDA 0.001f
#define LUB  32            // LU block size (multiple of 16, divides S)

typedef __attribute__((ext_vector_type(2))) float v2f;
typedef __attribute__((ext_vector_type(8))) float v8f;

// ---------------------------------------------------------------------------
// Workspace layout (floats):
//   M    : S*64      row-major [w^2 (32) | -2 w^2 x_sv (32)]
//   C    : S         sum_f w^2 x_sv^2
//   Psv  : 64*S      rows: [x_sv^2 (32) | x_sv (32)], column j = SV j
//   Ptr  : 64*N      same for x_train columns
//   Gt   : S*S       (K_sv + lambda I)^T  -> LU-factored in place
//   beta : 1
//   r    : S         y_sv - beta
//   z    : S         alpha (solution of Gt z = r)
// ---------------------------------------------------------------------------
#define OFF_M    0
#define OFF_C    (OFF_M   + S*64)
#define OFF_PSV  (OFF_C   + S)
#define OFF_PTR  (OFF_PSV + 64*S)
#define OFF_GT   (OFF_PTR + 64*N)
#define OFF_BETA (OFF_GT  + S*S)
#define OFF_R    (OFF_BETA + 1)
#define OFF_Z    (OFF_R   + S)

// ----------------------------- small prep kernels --------------------------

__global__ void k_beta(const float* __restrict__ ysv, float* __restrict__ beta) {
  __shared__ float red[256];
  float s = 0.f;
  for (int i = threadIdx.x; i < S; i += 256) s += ysv[i];
  red[threadIdx.x] = s; __syncthreads();
  for (int o = 128; o > 0; o >>= 1) {
    if ((int)threadIdx.x < o) red[threadIdx.x] += red[threadIdx.x + o];
    __syncthreads();
  }
  if (threadIdx.x == 0) *beta = red[0] * (1.0f / (float)S);
}

__global__ void k_resid(const float* __restrict__ ysv, const float* __restrict__ beta,
                        float* __restrict__ r) {
  int i = blockIdx.x * 256 + threadIdx.x;
  if (i < S) r[i] = ysv[i] - *beta;
}

// per-SV GEMM row M[s][0:64] and constant C[s]
__global__ void k_prep_sv(const float* __restrict__ xsv, const float* __restrict__ w,
                          float* __restrict__ M, float* __restrict__ C) {
  int s = blockIdx.x * 256 + threadIdx.x;
  if (s >= S) return;
  float c = 0.f;
  #pragma unroll
  for (int f = 0; f < F; ++f) {
    float wv = w[f * S + s];
    float xv = xsv[f * S + s];
    float w2 = wv * wv;
    M[s * 64 + f]      = w2;
    M[s * 64 + 32 + f] = -2.f * w2 * xv;
    c += w2 * xv * xv;
  }
  C[s] = c;
}

// column features P (64 x ncols row-major): rows 0..31 = x^2, rows 32..63 = x
__global__ void k_prep_cols(const float* __restrict__ x, float* __restrict__ P, int ncols) {
  int t = blockIdx.x * 256 + threadIdx.x;
  if (t >= F * ncols) return;
  int f = t / ncols, j = t - f * ncols;
  float xv = x[f * ncols + j];
  P[f * ncols + j]        = xv * xv;
  P[(F + f) * ncols + j]  = xv;
}

// ----------------------------- WMMA tile core -------------------------------
// 16x16 tile of  D2 = M(16x64) * P(64x16)  via 16 chained V_WMMA_F32_16X16X4_F32.
// A-frag: lane L -> row m=L&15, K pair base kb=(L>>4)*2 (ISA 7.12.2, 32-bit A 16x4).
// B-frag: lane L -> col n=L&15, same K pair base (row-striped across lanes).
// D:      lane L -> col n=L&15, VGPR r -> row m=r+(L>>4)*8.
__device__ __forceinline__ v8f wmma_d2_tile(const float* __restrict__ M,
                                            const float* __restrict__ P,
                                            int i0, int j0, int ldP, int lane) {
  const int mn = lane & 15;
  const int kb = (lane >> 4) * 2;
  const float* Mrow = M + (size_t)(i0 + mn) * 64;
  const float* Pcol = P + j0 + mn;
  v8f acc = {};
  #pragma unroll
  for (int kk = 0; kk < 16; ++kk) {
    const int k = kk * 4 + kb;
    v2f a, b;
    a.x = Mrow[k];
    a.y = Mrow[k + 1];
    b.x = Pcol[(size_t)k * ldP];
    b.y = Pcol[(size_t)(k + 1) * ldP];
    acc = __builtin_amdgcn_wmma_f32_16x16x4_f32(
        /*neg_a=*/false, a, /*neg_b=*/false, b,
        /*c_mod=*/(short)0, acc, /*reuse_a=*/false, /*reuse_b=*/false);
  }
  return acc;
}

// Gram matrix, stored TRANSPOSED with ridge term: Gt[j][i] = K_sv[i][j] + lam*(i==j)
__global__ void k_ksv(const float* __restrict__ M, const float* __restrict__ P,
                      const float* __restrict__ C, float* __restrict__ Gt) {
  const int lane = threadIdx.x & 31;
  const int wave = threadIdx.x >> 5;
  const int j0 = (blockIdx.x * 8 + wave) * 16;
  const int i0 = blockIdx.y * 16;
  v8f acc = wmma_d2_tile(M, P, i0, j0, S, lane);
  const int n  = lane & 15;
  const int mh = (lane >> 4) * 8;
  #pragma unroll
  for (int r2 = 0; r2 < 8; ++r2) {
    const int gi = i0 + r2 + mh;
    const int gj = j0 + n;
    float d2 = acc[r2] + C[gi];
    float v  = (d2 > 0.f) ? expf(-sqrtf(d2)) : 1.0f;   // exp(0)=1 branch of safe_norm
    if (gi == gj) v += LAMDA;
    Gt[(size_t)gj * S + gi] = v;
  }
}

// Fused: build K_train tile with WMMA, weight by alpha, accumulate y per column.
__global__ void k_pred(const float* __restrict__ M, const float* __restrict__ P,
                       const float* __restrict__ C, const float* __restrict__ z,
                       const float* __restrict__ beta, float* __restrict__ out) {
  const int lane = threadIdx.x & 31;
  const int wave = threadIdx.x >> 5;
  const int j0 = (blockIdx.x * 8 + wave) * 16;
  const int n  = lane & 15;
  const int mh = (lane >> 4) * 8;
  float ysum = 0.f;
  for (int i0 = 0; i0 < S; i0 += 16) {
    __builtin_prefetch(M + (size_t)(i0 + 16 + (lane & 15)) * 64, 0, 0);
    v8f acc = wmma_d2_tile(M, P, i0, j0, N, lane);
    #pragma unroll
    for (int r2 = 0; r2 < 8; ++r2) {
      const int gi = i0 + r2 + mh;
      float d2 = acc[r2] + C[gi];
      float kv = (d2 > 0.f) ? expf(-sqrtf(d2)) : 1.0f;
      ysum += z[gi] * kv;
    }
  }
  ysum += __shfl_xor(ysum, 16, 32);   // combine the two half-wave row groups
  if (lane < 16) out[j0 + n] = ysum + *beta;
}

// ----------------------------- blocked LU (unpivoted) ------------------------

// Factor the panel A[k:S, k:k+LUB] in place. Single workgroup, serial over the
// LUB columns; __syncthreads orders the global-memory phases within the block.
__global__ void k_panel(float* __restrict__ A, int k) {
  for (int c = 0; c < LUB; ++c) {
    const int col = k + c;
    const float piv = A[(size_t)col * S + col];
    for (int i = col + 1 + threadIdx.x; i < S; i += 256)
      A[(size_t)i * S + col] /= piv;
    __syncthreads();
    const int rows = S - col - 1;          // rows below pivot
    const int cols = LUB - 1 - c;          // remaining panel columns
    const int tot  = rows * cols;
    for (int idx = threadIdx.x; idx < tot; idx += 256) {
      const int i = col + 1 + idx / cols;
      const int j = col + 1 + idx % cols;
      A[(size_t)i * S + j] -= A[(size_t)i * S + col] * A[(size_t)col * S + j];
    }
    __syncthreads();
  }
}

// U12 = L11^{-1} * A12 : one thread per trailing column, 32-row triangular
// solve held entirely in registers.
__global__ void k_tsolve(float* __restrict__ A, int k) {
  const int j = k + LUB + blockIdx.x * 256 + threadIdx.x;
  if (j >= S) return;
  float u[LUB];
  #pragma unroll
  for (int r2 = 0; r2 < LUB; ++r2) {
    float s = A[(size_t)(k + r2) * S + j];
    #pragma unroll
    for (int c = 0; c < LUB; ++c)
      if (c < r2) s -= A[(size_t)(k + r2) * S + (k + c)] * u[c];
    u[r2] = s;
    A[(size_t)(k + r2) * S + j] = s;
  }
}

// Trailing update A22 -= L21 * U12 as a K=LUB WMMA GEMM.
// C-fragment seeded from A22; A-fragment negated at load (f32 WMMA has no A-neg).
__global__ void k_trail(float* __restrict__ A, int k) {
  const int lane = threadIdx.x & 31;
  const int wave = threadIdx.x >> 5;
  const int base = k + LUB;
  const int j0 = base + (blockIdx.x * 8 + wave) * 16;
  if (j0 >= S) return;                       // wave-uniform guard: EXEC stays all-1s
  const int i0 = base + blockIdx.y * 16;
  const int mn = lane & 15;
  const int kb = (lane >> 4) * 2;
  const int n  = lane & 15;
  const int mh = (lane >> 4) * 8;

  v8f acc;
  #pragma unroll
  for (int r2 = 0; r2 < 8; ++r2)
    acc[r2] = A[(size_t)(i0 + r2 + mh) * S + j0 + n];

  const float* L21row = A + (size_t)(i0 + mn) * S + k;   // A[i0+m][k + *]
  const float* U12col = A + (size_t)k * S + j0 + mn;     // A[k + *][j0+n]
  #pragma unroll
  for (int kk = 0; kk < LUB / 4; ++kk) {
    const int kc = kk * 4 + kb;
    v2f a, b;
    a.x = -L21row[kc];
    a.y = -L21row[kc + 1];
    b.x = U12col[(size_t)kc * S];
    b.y = U12col[(size_t)(kc + 1) * S];
    acc = __builtin_amdgcn_wmma_f32_16x16x4_f32(
        false, a, false, b, (short)0, acc, false, false);
  }

  #pragma unroll
  for (int r2 = 0; r2 < 8; ++r2)
    A[(size_t)(i0 + r2 + mh) * S + j0 + n] = acc[r2];
}

// L y = b (unit lower triangular), solution kept in LDS
__global__ void k_fwd(const float* __restrict__ A, const float* __restrict__ b,
                      float* __restrict__ y) {
  __shared__ float red[256];
  __shared__ float yv[S];
  for (int i = 0; i < S; ++i) {
    float s = 0.f;
    for (int j = threadIdx.x; j < i; j += 256) s += A[(size_t)i * S + j] * yv[j];
    red[threadIdx.x] = s; __syncthreads();
    for (int o = 128; o > 0; o >>= 1) {
      if ((int)threadIdx.x < o) red[threadIdx.x] += red[threadIdx.x + o];
      __syncthreads();
    }
    if (threadIdx.x == 0) yv[i] = b[i] - red[0];
    __syncthreads();
  }
  for (int i = threadIdx.x; i < S; i += 256) y[i] = yv[i];
}

// U z = y
__global__ void k_bwd(const float* __restrict__ A, const float* __restrict__ b,
                      float* __restrict__ z) {
  __shared__ float red[256];
  __shared__ float zv[S];
  for (int i = S - 1; i >= 0; --i) {
    float s = 0.f;
    for (int j = i + 1 + threadIdx.x; j < S; j += 256) s += A[(size_t)i * S + j] * zv[j];
    red[threadIdx.x] = s; __syncthreads();
    for (int o = 128; o > 0; o >>= 1) {
      if ((int)threadIdx.x < o) red[threadIdx.x] += red[threadIdx.x + o];
      __syncthreads();
    }
    if (threadIdx.x == 0) zv[i] = (b[i] - red[0]) / A[(size_t)i * S + i];
    __syncthreads();
  }
  for (int i = threadIdx.x; i < S; i += 256) z[i] = zv[i];
}

// ----------------------------- host driver ----------------------------------

extern "C" void kernel_launch(void* const* d_in, const int* in_sizes, int n_in,
                              void* d_out, int out_size, void* d_ws, size_t ws_size,
                              hipStream_t stream) {
  const float* x_train = (const float*)d_in[0];  // (F, N)
  const float* x_sv    = (const float*)d_in[1];  // (F, S)
  const float* y_sv    = (const float*)d_in[2];  // (S,)
  const float* weight  = (const float*)d_in[3];  // (F, S)
  float* out = (float*)d_out;                    // (1, N) fp32

  float* ws   = (float*)d_ws;
  float* M    = ws + OFF_M;
  float* C    = ws + OFF_C;
  float* Psv  = ws + OFF_PSV;
  float* Ptr  = ws + OFF_PTR;
  float* Gt   = ws + OFF_GT;
  float* beta = ws + OFF_BETA;
  float* r    = ws + OFF_R;
  float* z    = ws + OFF_Z;

  // beta, residual
  k_beta <<<1, 256, 0, stream>>>(y_sv, beta);
  k_resid<<<(S + 255) / 256, 256, 0, stream>>>(y_sv, beta, r);

  // GEMM-form precomputation
  k_prep_sv  <<<(S + 255) / 256, 256, 0, stream>>>(x_sv, weight, M, C);
  k_prep_cols<<<(F * S + 255) / 256, 256, 0, stream>>>(x_sv, Psv, S);
  k_prep_cols<<<(F * N + 255) / 256, 256, 0, stream>>>(x_train, Ptr, N);

  // Gt = (K_sv + lam I)^T  via WMMA tiles (8 waves/block, 16x16 tile per wave)
  k_ksv<<<dim3(S / 128, S / 16), 256, 0, stream>>>(M, Psv, C, Gt);

  // blocked unpivoted LU of Gt in place: panel / tri-solve / WMMA trailing GEMM
  for (int k = 0; k < S; k += LUB) {
    k_panel<<<1, 256, 0, stream>>>(Gt, k);
    const int n2 = S - k - LUB;                 // trailing rows == trailing cols
    if (n2 > 0) {
      k_tsolve<<<dim3((n2 + 255) / 256), 256, 0, stream>>>(Gt, k);
      k_trail <<<dim3((n2 + 127) / 128, n2 / 16), 256, 0, stream>>>(Gt, k);
    }
  }

  // Gt z = r  (alpha = z)
  k_fwd<<<1, 256, 0, stream>>>(Gt, r, z);   // L y = r
  k_bwd<<<1, 256, 0, stream>>>(Gt, z, z);   // U z = y (aliasing safe: zv lives in LDS)

  // fused K_train build + alpha dot + beta
  k_pred<<<dim3(N / 128), 256, 0, stream>>>(M, Ptr, C, z, beta, out);
}